// WaveletBlock3D_85555748536332
// MI455X (gfx1250) — compile-verified
//
#include <hip/hip_runtime.h>

// WaveletBlock3D for MI455X (gfx1250), f32 end-to-end.
// GEMMs on V_WMMA_F32_16X16X4_F32; weight panels staged to LDS via the
// Tensor Data Mover (TENSOR_LOAD_TO_LDS + s_wait_tensorcnt) when available.

typedef __attribute__((ext_vector_type(2))) float v2f;
typedef __attribute__((ext_vector_type(8))) float v8f;
typedef __attribute__((ext_vector_type(4))) unsigned int u32x4;
typedef __attribute__((ext_vector_type(8))) int i32x8;
typedef __attribute__((ext_vector_type(4))) int i32x4;

#define WB_B   2
#define WB_C   32
#define WB_S   (64*64*64)   /* 262144 full-res voxels per batch */
#define WB_DW  128
#define WB_S2  (32*32*32)   /* 32768 half-res voxels per batch */
#define EPSV   1e-6f

#if defined(__has_builtin)
#if __has_builtin(__builtin_amdgcn_tensor_load_to_lds) && \
    __has_builtin(__builtin_amdgcn_s_wait_tensorcnt)
#define WB_USE_TDM 1
#endif
#endif
#ifndef WB_USE_TDM
#define WB_USE_TDM 0
#endif

static __device__ __forceinline__ v8f wmma_f32_4(v2f a, v2f b, v8f c) {
  // 8 args: (neg_a, A, neg_b, B, c_mod, C, reuse_a, reuse_b)
  return __builtin_amdgcn_wmma_f32_16x16x4_f32(false, a, false, b, (short)0, c,
                                               false, false);
}

#if WB_USE_TDM
// TDM load of a 2D [rows x cols_dw] f32 tile from global to LDS with one pad
// DWORD inserted every (2<<pad_code) DWORDs (LDS row stride = cols+1).
// D# layout per CDNA5 ISA ch.8: group0 = {ctrl, lds_addr, global_addr, type},
// group1 = {flags/pad, tensor dims, tile dims, dim0 stride}; groups 2/3 zero
// (2D tensor). Issue from ONE wave only; TDM ignores EXEC.
static __device__ __forceinline__ void tdm_load_2d_padded(
    unsigned lds_off, const void* gptr, unsigned rows, unsigned cols_dw,
    unsigned pad_code) {
  unsigned long long ga = (unsigned long long)(size_t)gptr;
  u32x4 g0;
  g0[0] = 1u;                                         // count=1 (valid user D#)
  g0[1] = lds_off;                                    // lds_addr (bytes)
  g0[2] = (unsigned)(ga & 0xFFFFFFFFu);               // global_addr[31:0]
  g0[3] = (unsigned)((ga >> 32) & 0x01FFFFFFu)        // global_addr[56:32]
          | (2u << 30);                               // type=2 ("image")
  i32x8 g1;
  g1[0] = (int)((2u << 16)                            // data_size = 4B
                | (1u << 20)                          // pad_enable
                | (pad_code << 22)                    // pad_interval
                | (0u << 25));                        // pad_amount = 1 DWORD
  g1[1] = (int)((cols_dw & 0xFFFFu) << 16);           // tensor_dim0 lo16
  g1[2] = (int)(((cols_dw >> 16) & 0xFFFFu)           // tensor_dim0 hi16
                | ((rows & 0xFFFFu) << 16));          // tensor_dim1 lo16
  g1[3] = (int)(((rows >> 16) & 0xFFFFu)              // tensor_dim1 hi16
                | ((cols_dw & 0xFFFFu) << 16));       // tile_dim0
  g1[4] = (int)(rows & 0xFFFFu);                      // tile_dim1 (tile_dim2=0)
  g1[5] = (int)cols_dw;                               // tensor_dim0_stride lo
  g1[6] = 0;
  g1[7] = 0;
  i32x4 g2 = {0, 0, 0, 0}, g3 = {0, 0, 0, 0};
#if defined(__clang_major__) && (__clang_major__ >= 23)
  i32x8 g4 = {0, 0, 0, 0, 0, 0, 0, 0};
  __builtin_amdgcn_tensor_load_to_lds(g0, g1, g2, g3, g4, 0);
#else
  __builtin_amdgcn_tensor_load_to_lds(g0, g1, g2, g3, 0);
#endif
}
#endif // WB_USE_TDM

// ---------------------------------------------------------------- kernel 1
// Channel LayerNorm over C=32 for each voxel (channel stride = WB_S).
__global__ void k_ln1(const float* __restrict__ inp, const float* __restrict__ w,
                      const float* __restrict__ b, float* __restrict__ out) {
  int v = blockIdx.x * blockDim.x + threadIdx.x;
  if (v >= WB_B * WB_S) return;
  int bb = v / WB_S, sp = v % WB_S;
  size_t base = (size_t)bb * WB_C * WB_S + sp;
  float s = 0.f, s2 = 0.f;
  for (int c = 0; c < WB_C; ++c) {
    float x = inp[base + (size_t)c * WB_S];
    s += x; s2 += x * x;
  }
  float mu  = s  * (1.0f / WB_C);
  float var = s2 * (1.0f / WB_C) - mu * mu;
  float r = rsqrtf(var + EPSV);
  for (int c = 0; c < WB_C; ++c) {
    float x = inp[base + (size_t)c * WB_S];
    out[base + (size_t)c * WB_S] = (x - mu) * r * w[c] + b[c];
  }
}

// ---------------------------------------------------------------- kernel 2
// Depthwise 4x4x4 wavelet conv, stride 2, pad 1. Each block shares one output
// channel -> stage its 64 taps in LDS.
__global__ void k_wave(const float* __restrict__ x1, const float* __restrict__ wk,
                       float* __restrict__ out) {
  __shared__ float kw[64];
  int flat0 = blockIdx.x * 256;
  int oc = (flat0 / WB_S2) & 127;
  if (threadIdx.x < 64) kw[threadIdx.x] = wk[oc * 64 + threadIdx.x];
  __syncthreads();
  int flat = flat0 + threadIdx.x;
  int bb  = flat / (WB_DW * WB_S2);
  int sp2 = flat & (WB_S2 - 1);
  int d2 = sp2 >> 10, h2 = (sp2 >> 5) & 31, w2 = sp2 & 31;
  int cin = oc >> 2;                       // feature_group_count = C
  const float* src = x1 + ((size_t)bb * WB_C + cin) * WB_S;
  float acc = 0.f;
  for (int kd = 0; kd < 4; ++kd) {
    int id = 2 * d2 + kd - 1;
    if ((unsigned)id >= 64u) continue;
    for (int kh = 0; kh < 4; ++kh) {
      int ih = 2 * h2 + kh - 1;
      if ((unsigned)ih >= 64u) continue;
      const float* row = src + ((size_t)id * 64 + ih) * 64;
      __builtin_prefetch(row + 128, 0, 0);   // global_prefetch_b8 hint
      for (int kwi = 0; kwi < 4; ++kwi) {
        int iw = 2 * w2 + kwi - 1;
        if ((unsigned)iw >= 64u) continue;
        acc += kw[(kd * 4 + kh) * 4 + kwi] * row[iw];
      }
    }
  }
  out[flat] = acc;
}

// ---------------------------------------------------------------- kernel 3
// LayerNorm over 128 channels at half resolution, in place.
__global__ void k_lna(float* __restrict__ x, const float* __restrict__ w,
                      const float* __restrict__ b) {
  int v = blockIdx.x * blockDim.x + threadIdx.x;
  if (v >= WB_B * WB_S2) return;
  int bb = v / WB_S2, sp = v & (WB_S2 - 1);
  size_t base = (size_t)bb * WB_DW * WB_S2 + sp;
  float s = 0.f, s2 = 0.f;
  for (int c = 0; c < WB_DW; ++c) {
    float t = x[base + (size_t)c * WB_S2]; s += t; s2 += t * t;
  }
  float mu  = s  * (1.0f / WB_DW);
  float var = s2 * (1.0f / WB_DW) - mu * mu;
  float r = rsqrtf(var + EPSV);
  for (int c = 0; c < WB_DW; ++c) {
    float t = x[base + (size_t)c * WB_S2];
    x[base + (size_t)c * WB_S2] = (t - mu) * r * w[c] + b[c];
  }
}

// ---------------------------------------------------------------- kernel 4
// Per-(b,c) spatial max (gate input), block reduction.
__global__ void k_max(const float* __restrict__ x, float* __restrict__ m) {
  __shared__ float red[256];
  int bc = blockIdx.x;                    // 0..255 = (b,c)
  const float* src = x + (size_t)bc * WB_S2;
  float mx = -3.4e38f;
  for (int i = threadIdx.x; i < WB_S2; i += 256) mx = fmaxf(mx, src[i]);
  red[threadIdx.x] = mx; __syncthreads();
  for (int o = 128; o > 0; o >>= 1) {
    if (threadIdx.x < o) red[threadIdx.x] = fmaxf(red[threadIdx.x], red[threadIdx.x + o]);
    __syncthreads();
  }
  if (threadIdx.x == 0) m[bc] = red[0];
}

// ---------------------------------------------------------------- kernel 5
// gate[b][o] = sigmoid(sca_w[o,:] . m[b,:] + sca_b[o])   (tiny 128x128 matvec)
__global__ void k_gate(const float* __restrict__ m, const float* __restrict__ w,
                       const float* __restrict__ b, float* __restrict__ g) {
  int bb = blockIdx.x, o = threadIdx.x;
  float acc = b[o];
  for (int c = 0; c < WB_DW; ++c) acc += w[o * WB_DW + c] * m[bb * WB_DW + c];
  g[bb * WB_DW + o] = 1.0f / (1.0f + __expf(-acc));
}

// ---------------------------------------------------------------- kernel 6
// Fused: trilinear x2 upsample gather -> LDS panel, gate folded into weights,
// conv3 GEMM (M=32, K=128, N=64 voxels/block) via v_wmma_f32_16x16x4_f32,
// + conv3 bias + beta residual. Writes y. conv3_w panel arrives via TDM.
__global__ void __launch_bounds__(256)
k_up_gate_conv3(const float* __restrict__ x2, const float* __restrict__ gate,
                const float* __restrict__ c3w, const float* __restrict__ c3b,
                const float* __restrict__ inp, const float* __restrict__ beta,
                float* __restrict__ y) {
  __shared__ float As[32 * 129];   // weights, gate folded in after barrier
  __shared__ float Bs[128 * 65];   // upsampled activations [c][vox]
  const int tid = threadIdx.x;
  const int vb  = blockIdx.x * 64;     // 64 voxels per block (one w-row)
  const int bb  = vb / WB_S;
  const int spb = vb % WB_S;

#if WB_USE_TDM
  if (tid < 32) {                      // wave 0 issues the DMA
    // 32 rows x 128 DWORDs, 1 pad DWORD per 128 (code 6) -> LDS stride 129
    tdm_load_2d_padded((unsigned)(size_t)As, c3w, 32, 128, 6);
    __builtin_amdgcn_s_wait_tensorcnt(0);
  }
#else
  for (int i = tid; i < 32 * 128; i += 256)
    As[(i >> 7) * 129 + (i & 127)] = c3w[i];
#endif

  // per-thread voxel v = tid & 63; precompute 8 trilinear taps once
  int v = tid & 63;
  int sp = spb + v;
  int crd[3] = { sp >> 12, (sp >> 6) & 63, sp & 63 };
  int i0[3], i1[3]; float f1[3];
  for (int a = 0; a < 3; ++a) {
    float cc = crd[a] * 0.5f - 0.25f;     // (out+0.5)/2 - 0.5
    float fl = floorf(cc);
    float t  = cc - fl;
    int j0 = (int)fl, j1 = j0 + 1;
    j0 = min(max(j0, 0), 31); j1 = min(max(j1, 0), 31);
    i0[a] = j0; i1[a] = j1; f1[a] = t;
  }
  int nb[8]; float wt[8];
  for (int k = 0; k < 8; ++k) {
    int dz = (k >> 2) & 1, dy = (k >> 1) & 1, dx = k & 1;
    int zi = dz ? i1[0] : i0[0], yi = dy ? i1[1] : i0[1], xi = dx ? i1[2] : i0[2];
    nb[k] = (zi * 32 + yi) * 32 + xi;
    wt[k] = (dz ? f1[0] : 1.f - f1[0]) * (dy ? f1[1] : 1.f - f1[1]) *
            (dx ? f1[2] : 1.f - f1[2]);
  }
  const float* xb = x2 + (size_t)bb * WB_DW * WB_S2;
  for (int c = tid >> 6; c < WB_DW; c += 4) {
    const float* src = xb + (size_t)c * WB_S2;
    if (c + 4 < WB_DW) __builtin_prefetch(src + 4 * WB_S2 + nb[0], 0, 0);
    float val = 0.f;
    for (int k = 0; k < 8; ++k) val += wt[k] * src[nb[k]];
    Bs[c * 65 + v] = val;
  }
  __syncthreads();

  // fold gate into the weight panel (TDM delivered raw conv3_w)
  for (int i = tid; i < 32 * 128; i += 256) {
    int o = i >> 7, c = i & 127;
    As[o * 129 + c] *= gate[bb * 128 + c];
  }
  __syncthreads();

  // 8 waves -> 8 subtiles (mt in 0..1, nt in 0..3). EXEC all ones here.
  int wv = tid >> 5, lane = tid & 31, l = lane & 15, half = lane >> 4;
  int mt = wv >> 2, nt = wv & 3;
  v8f acc = {};
  for (int k = 0; k < 128; k += 4) {
    int kk = k + 2 * half;                 // A: lanes0-15 K{0,1}, 16-31 K{2,3}
    v2f a, b2;
    a.x  = As[(mt * 16 + l) * 129 + kk];
    a.y  = As[(mt * 16 + l) * 129 + kk + 1];
    b2.x = Bs[kk * 65 + nt * 16 + l];
    b2.y = Bs[(kk + 1) * 65 + nt * 16 + l];
    acc = wmma_f32_4(a, b2, acc);
  }
  for (int r = 0; r < 8; ++r) {
    int o  = mt * 16 + r + 8 * half;       // C/D: M = vgpr + 8*half, N = l
    int vv = nt * 16 + l;
    size_t addr = ((size_t)bb * WB_C + o) * WB_S + (spb + vv);
    float xval = acc[r] + c3b[o];
    y[addr] = inp[addr] + beta[o] * xval;
  }
}

// ---------------------------------------------------------------- kernel 7
// Fused FFN: LN2 -> conv4 GEMM (64x32) -> gated split product -> conv5 GEMM
// (32x32) -> gamma residual. GEMMs on WMMA, weight panels via TDM.
__global__ void __launch_bounds__(256)
k_ffn(const float* __restrict__ y, const float* __restrict__ n2w,
      const float* __restrict__ n2b, const float* __restrict__ c4w,
      const float* __restrict__ c4b, const float* __restrict__ c5w,
      const float* __restrict__ c5b, const float* __restrict__ gamma,
      float* __restrict__ out) {
  __shared__ float Xs[32 * 65];
  __shared__ float Hs[64 * 65];
  __shared__ float W4[64 * 33];
  __shared__ float W5[32 * 33];
  const int tid = threadIdx.x;
  const int vb  = blockIdx.x * 64;
  const int bb  = vb / WB_S;
  const int spb = vb % WB_S;

#if WB_USE_TDM
  if (tid < 32) {                      // wave 0 issues both DMAs
    // rows x 32 DWORDs, 1 pad DWORD per 32 (code 4) -> LDS stride 33
    tdm_load_2d_padded((unsigned)(size_t)W4, c4w, 64, 32, 4);
    tdm_load_2d_padded((unsigned)(size_t)W5, c5w, 32, 32, 4);
    __builtin_amdgcn_s_wait_tensorcnt(0);
  }
#else
  for (int i = tid; i < 64 * 32; i += 256) W4[(i >> 5) * 33 + (i & 31)] = c4w[i];
  for (int i = tid; i < 32 * 32; i += 256) W5[(i >> 5) * 33 + (i & 31)] = c5w[i];
#endif

  if (tid < 64) {
    int v = tid;
    size_t base = (size_t)bb * WB_C * WB_S + spb + v;
    float s = 0.f, s2 = 0.f;
    for (int c = 0; c < 32; ++c) {
      float t = y[base + (size_t)c * WB_S]; s += t; s2 += t * t;
    }
    float mu = s * (1.f / 32.f), var = s2 * (1.f / 32.f) - mu * mu;
    float r = rsqrtf(var + EPSV);
    for (int c = 0; c < 32; ++c) {
      float t = y[base + (size_t)c * WB_S];
      Xs[c * 65 + v] = (t - mu) * r * n2w[c] + n2b[c];
    }
  }
  __syncthreads();

  int wv = tid >> 5, lane = tid & 31, l = lane & 15, half = lane >> 4;
  int nt = wv & 3;
  // GEMM1: M=64 -> 16 subtiles, 2 per wave
  for (int i = 0; i < 2; ++i) {
    int mt = (wv >> 2) + 2 * i;
    v8f acc = {};
    for (int k = 0; k < 32; k += 4) {
      int kk = k + 2 * half;
      v2f a, b2;
      a.x  = W4[(mt * 16 + l) * 33 + kk];
      a.y  = W4[(mt * 16 + l) * 33 + kk + 1];
      b2.x = Xs[kk * 65 + nt * 16 + l];
      b2.y = Xs[(kk + 1) * 65 + nt * 16 + l];
      acc = wmma_f32_4(a, b2, acc);
    }
    for (int r = 0; r < 8; ++r) {
      int o = mt * 16 + r + 8 * half;
      Hs[o * 65 + nt * 16 + l] = acc[r] + c4b[o];
    }
  }
  __syncthreads();
  // gated split product g = h[0:32] * h[32:64], reuse Xs
  for (int i = tid; i < 32 * 64; i += 256) {
    int c = i >> 6, v = i & 63;
    Xs[c * 65 + v] = Hs[c * 65 + v] * Hs[(c + 32) * 65 + v];
  }
  __syncthreads();
  // GEMM2: M=32 -> 8 subtiles, 1 per wave
  {
    int mt = wv >> 2;
    v8f acc = {};
    for (int k = 0; k < 32; k += 4) {
      int kk = k + 2 * half;
      v2f a, b2;
      a.x  = W5[(mt * 16 + l) * 33 + kk];
      a.y  = W5[(mt * 16 + l) * 33 + kk + 1];
      b2.x = Xs[kk * 65 + nt * 16 + l];
      b2.y = Xs[(kk + 1) * 65 + nt * 16 + l];
      acc = wmma_f32_4(a, b2, acc);
    }
    for (int r = 0; r < 8; ++r) {
      int o  = mt * 16 + r + 8 * half;
      int vv = nt * 16 + l;
      size_t addr = ((size_t)bb * WB_C + o) * WB_S + spb + vv;
      out[addr] = y[addr] + gamma[o] * (acc[r] + c5b[o]);
    }
  }
}

// ---------------------------------------------------------------- launch
extern "C" void kernel_launch(void* const* d_in, const int* in_sizes, int n_in,
                              void* d_out, int out_size, void* d_ws, size_t ws_size,
                              hipStream_t stream) {
  const float* inp   = (const float*)d_in[0];
  const float* n1w   = (const float*)d_in[1];
  const float* n1b   = (const float*)d_in[2];
  const float* wker  = (const float*)d_in[3];
  const float* naw   = (const float*)d_in[4];
  const float* nab   = (const float*)d_in[5];
  const float* scaw  = (const float*)d_in[6];
  const float* scab  = (const float*)d_in[7];
  const float* c3w   = (const float*)d_in[8];
  const float* c3b   = (const float*)d_in[9];
  const float* n2w   = (const float*)d_in[10];
  const float* n2b   = (const float*)d_in[11];
  const float* c4w   = (const float*)d_in[12];
  const float* c4b   = (const float*)d_in[13];
  const float* c5w   = (const float*)d_in[14];
  const float* c5b   = (const float*)d_in[15];
  const float* beta  = (const float*)d_in[16];
  const float* gamma = (const float*)d_in[17];
  float* out = (float*)d_out;

  float* ws   = (float*)d_ws;
  float* x1y  = ws;                                   // 16,777,216 f (x1, then y)
  float* xw   = ws + (size_t)WB_B * WB_C * WB_S;      //  8,388,608 f (wavelet/LNa)
  float* mbuf = xw + (size_t)WB_B * WB_DW * WB_S2;    //        256 f
  float* gbuf = mbuf + 256;                           //        256 f

  k_ln1 <<<(WB_B * WB_S) / 256,        256, 0, stream>>>(inp, n1w, n1b, x1y);
  k_wave<<<(WB_B * WB_DW * WB_S2) / 256, 256, 0, stream>>>(x1y, wker, xw);
  k_lna <<<(WB_B * WB_S2) / 256,       256, 0, stream>>>(xw, naw, nab);
  k_max <<<WB_B * WB_DW,               256, 0, stream>>>(xw, mbuf);
  k_gate<<<WB_B,                     WB_DW, 0, stream>>>(mbuf, scaw, scab, gbuf);
  k_up_gate_conv3<<<(WB_B * WB_S) / 64, 256, 0, stream>>>(xw, gbuf, c3w, c3b,
                                                          inp, beta, x1y);
  k_ffn <<<(WB_B * WB_S) / 64,         256, 0, stream>>>(x1y, n2w, n2b, c4w, c4b,
                                                         c5w, c5b, gamma, out);
}